// CustomMLPLayer_20572893348634
// MI455X (gfx1250) — compile-verified
//
#include <hip/hip_runtime.h>

// ---- problem constants (from reference) ----
#define D_MODEL 4096
#define D_FF    11008
#define S_TOK   2048
#define K_TOK   2201        // per-token top-k
#define K_CORE  2201        // core neurons
#define NSPLIT  2201        // model_neurons prefix
#define TARGET  4403        // selected neurons
#define REMN    (D_FF - NSPLIT)   // 8807

typedef __attribute__((ext_vector_type(2))) float v2f;
typedef __attribute__((ext_vector_type(4))) float v4f;
typedef __attribute__((ext_vector_type(8))) float v8f;

// =====================================================================
// Kernel 1: true_value = x @ W^T   (M=2048, N=4096, K=11008, fp32 WMMA)
// Block tile 128x256, BK=32, double-buffered LDS, async global->LDS.
// 256 threads = 8 waves (2x4), wave tile 64x64 (4x4 WMMA accumulators).
// Both A and B staged k-major; every fragment is one aligned ds_load_b64.
// =====================================================================
#define BM 128
#define BN 256
#define BK 32
#define LDK 36            // padded row stride (144B: 16B-aligned rows, conflict-free)
#define USE_ASYNC 1       // 1: GLOBAL_LOAD_ASYNC_TO_LDS_B128 path (ASYNCcnt)

__device__ __forceinline__ void copy16_async(uint32_t lds_off, const float* g) {
#if USE_ASYNC
  // per-lane 16B copy global -> LDS, tracked by ASYNCcnt
  asm volatile("global_load_async_to_lds_b128 %0, %1, off"
               :: "v"(lds_off), "v"((unsigned long long)(uintptr_t)g)
               : "memory");
#else
  (void)lds_off; (void)g;
#endif
}

__global__ __launch_bounds__(256)
void gemm_f32_wmma(const float* __restrict__ X, const float* __restrict__ W,
                   float* __restrict__ Out) {
  __shared__ float As[2][BM * LDK];   // [m][k], k-major
  __shared__ float Bs[2][BN * LDK];   // [n][k], k-major

  const int tid  = threadIdx.x;
  const int lane = tid & 31;
  const int wave = tid >> 5;
  const int wm   = (wave >> 2) * 64;   // wave row offset in block tile (2 in M)
  const int wn   = (wave & 3) * 64;    // wave col offset in block tile (4 in N)

  const int gm0 = blockIdx.y * BM;
  const int gn0 = blockIdx.x * BN;

  const int hi2 = (lane >> 4) << 1;    // 0 (lanes 0-15) or 2 (lanes 16-31)
  const int nlo = lane & 15;

  v8f acc[4][4];
#pragma unroll
  for (int i = 0; i < 4; ++i)
#pragma unroll
    for (int j = 0; j < 4; ++j)
      acc[i][j] = (v8f){0.f,0.f,0.f,0.f,0.f,0.f,0.f,0.f};

  // ---- staging: 16B chunks; A: 4 chunks/thread, B: 8 chunks/thread ----
  auto stage = [&](int buf, int k0) {
#pragma unroll
    for (int r = 0; r < 4; ++r) {
      int q = r * 256 + tid;
      int row = q >> 3, c4 = (q & 7) << 2;
      const float* g = X + (size_t)(gm0 + row) * D_FF + k0 + c4;
      float* l = &As[buf][row * LDK + c4];
#if USE_ASYNC
      copy16_async((uint32_t)(uintptr_t)l, g);
#else
      *(v4f*)l = *(const v4f*)g;
#endif
    }
#pragma unroll
    for (int r = 0; r < 8; ++r) {
      int q = r * 256 + tid;
      int row = q >> 3, c4 = (q & 7) << 2;
      const float* g = W + (size_t)(gn0 + row) * D_FF + k0 + c4;
      float* l = &Bs[buf][row * LDK + c4];
#if USE_ASYNC
      copy16_async((uint32_t)(uintptr_t)l, g);
#else
      *(v4f*)l = *(const v4f*)g;
#endif
    }
  };

  auto wait_stage = [&]() {
#if USE_ASYNC
    asm volatile("s_wait_asynccnt 0x0" ::: "memory");
#endif
    __syncthreads();
  };

  const int NIT = D_FF / BK;    // 344
  stage(0, 0);
  wait_stage();

  for (int it = 0; it < NIT; ++it) {
    if (it + 1 < NIT) stage((it + 1) & 1, (it + 1) * BK);

    const float* pA = As[it & 1];
    const float* pB = Bs[it & 1];
#pragma unroll
    for (int kk = 0; kk < BK; kk += 4) {
      v2f af[4], bf[4];
#pragma unroll
      for (int mt = 0; mt < 4; ++mt)
        af[mt] = *(const v2f*)(pA + (wm + mt * 16 + nlo) * LDK + kk + hi2);
#pragma unroll
      for (int nt = 0; nt < 4; ++nt)
        bf[nt] = *(const v2f*)(pB + (wn + nt * 16 + nlo) * LDK + kk + hi2);
#pragma unroll
      for (int mt = 0; mt < 4; ++mt)
#pragma unroll
        for (int nt = 0; nt < 4; ++nt)
          acc[mt][nt] = __builtin_amdgcn_wmma_f32_16x16x4_f32(
              false, af[mt], false, bf[nt], (short)0, acc[mt][nt], false, false);
    }
    wait_stage();
  }

  // C/D 16x16 f32 layout: VGPR r, lanes 0-15 -> (M=r, N=lane), 16-31 -> (M=r+8)
  const int hi = lane >> 4;
#pragma unroll
  for (int mt = 0; mt < 4; ++mt)
#pragma unroll
    for (int nt = 0; nt < 4; ++nt)
#pragma unroll
      for (int r = 0; r < 8; ++r) {
        int m = gm0 + wm + mt * 16 + r + hi * 8;
        int n = gn0 + wn + nt * 16 + nlo;
        Out[(size_t)m * D_MODEL + n] = acc[mt][nt][r];
      }
}

// =====================================================================
// Kernel 2: zero scratch (counts + flags) every call (graph-replay safe)
// =====================================================================
__global__ void init_ws(int* __restrict__ counts, int* __restrict__ flags) {
  int i = blockIdx.x * blockDim.x + threadIdx.x;
  if (i < D_FF) { counts[i] = 0; flags[i] = 0; }
}

// =====================================================================
// Kernel 3: per-row exact top-K_TOK via 4-pass MSB radix select; votes
// =====================================================================
__global__ __launch_bounds__(256)
void topk_counts(const float* __restrict__ X, int* __restrict__ counts) {
  const int row = blockIdx.x;
  const float* xr = X + (size_t)row * D_FF;
  __shared__ int hist[256];
  __shared__ unsigned s_thresh;
  __shared__ int s_krem;
  __shared__ int s_eq;
  const int tid = threadIdx.x;

  unsigned prefix = 0, pmask = 0;
  int krem = K_TOK;
  for (int pass = 0; pass < 4; ++pass) {
    const int shift = 24 - pass * 8;
    hist[tid] = 0;
    __syncthreads();
    for (int i = tid; i < D_FF; i += 256) {
      unsigned u   = __float_as_uint(xr[i]);
      unsigned key = (u & 0x80000000u) ? ~u : (u | 0x80000000u); // order-preserving
      if ((key & pmask) == prefix)
        atomicAdd(&hist[(key >> shift) & 0xFF], 1);
    }
    __syncthreads();
    if (tid == 0) {
      int cum = 0, b = 255;
      for (; b > 0; --b) { cum += hist[b]; if (cum >= krem) break; }
      if (cum < krem) cum += hist[0];           // b == 0 fallthrough
      s_thresh = prefix | ((unsigned)b << shift);
      s_krem   = krem - (cum - hist[b]);
    }
    __syncthreads();
    prefix = s_thresh;
    krem   = s_krem;
    pmask  = 0xFFFFFFFFu << shift;
    __syncthreads();
  }
  if (tid == 0) s_eq = 0;
  __syncthreads();
  const unsigned T = prefix;
  for (int i = tid; i < D_FF; i += 256) {
    unsigned u   = __float_as_uint(xr[i]);
    unsigned key = (u & 0x80000000u) ? ~u : (u | 0x80000000u);
    if (key > T) {
      atomicAdd(&counts[i], 1);
    } else if (key == T) {                       // distinct keys: exactly krem of these
      int t = atomicAdd(&s_eq, 1);
      if (t < krem) atomicAdd(&counts[i], 1);
    }
  }
}

// =====================================================================
// Kernel 4: split flags (model_neurons[:NSPLIT])
// =====================================================================
__global__ void split_flags(const int* __restrict__ mn, int* __restrict__ flags) {
  int t = blockIdx.x * blockDim.x + threadIdx.x;
  if (t < NSPLIT) flags[mn[t]] = 1;
}

// =====================================================================
// Kernel 5: core = top-K_CORE of counts (ties -> ascending index); OR into flags
// =====================================================================
__global__ __launch_bounds__(1024)
void core_flags(const int* __restrict__ counts, int* __restrict__ flags) {
  __shared__ int chist[S_TOK + 1];
  __shared__ int s_c, s_need, sbase;
  __shared__ int sscan[1024];
  const int tid = threadIdx.x;

  for (int i = tid; i <= S_TOK; i += 1024) chist[i] = 0;
  __syncthreads();
  for (int i = tid; i < D_FF; i += 1024) atomicAdd(&chist[counts[i]], 1);
  __syncthreads();
  if (tid == 0) {
    int cum = 0, c = S_TOK;
    for (; c > 0; --c) { cum += chist[c]; if (cum >= K_CORE) break; }
    if (cum < K_CORE) cum += chist[0];
    s_c    = c;
    s_need = K_CORE - (cum - chist[c]);  // how many count==c entries to take
    sbase  = 0;
  }
  __syncthreads();
  const int cthr = s_c, need = s_need;

  for (int base = 0; base < D_FF; base += 1024) {
    int i = base + tid;
    int pred = 0;
    if (i < D_FF) {
      int cv = counts[i];
      if (cv > cthr) flags[i] = 1;
      pred = (cv == cthr) ? 1 : 0;
    }
    sscan[tid] = pred;
    __syncthreads();
    for (int off = 1; off < 1024; off <<= 1) {
      int v = (tid >= off) ? sscan[tid - off] : 0;
      __syncthreads();
      sscan[tid] += v;
      __syncthreads();
    }
    int excl = sscan[tid] - pred;
    int tot  = sscan[1023];
    if (pred && i < D_FF && (sbase + excl) < need) flags[i] = 1;
    __syncthreads();
    if (tid == 0) sbase += tot;
    __syncthreads();
  }
}

// =====================================================================
// Kernel 6: idx_all = sorted(union) ++ (rem model neurons not in union)[:TARGET]
// =====================================================================
__global__ __launch_bounds__(1024)
void build_idx(const int* __restrict__ flags, const int* __restrict__ mn,
               int* __restrict__ idx_all) {
  __shared__ int sscan[1024];
  __shared__ int sbase;
  const int tid = threadIdx.x;
  if (tid == 0) sbase = 0;
  __syncthreads();

  // phase 1: union members in ascending neuron order
  for (int base = 0; base < D_FF; base += 1024) {
    int i    = base + tid;
    int pred = (i < D_FF) ? flags[i] : 0;
    sscan[tid] = pred;
    __syncthreads();
    for (int off = 1; off < 1024; off <<= 1) {
      int v = (tid >= off) ? sscan[tid - off] : 0;
      __syncthreads();
      sscan[tid] += v;
      __syncthreads();
    }
    int pos = sbase + sscan[tid] - pred;
    if (pred && pos < TARGET) idx_all[pos] = i;
    __syncthreads();
    if (tid == 0) sbase += sscan[1023];
    __syncthreads();
  }
  // phase 2: remaining model neurons (not already in union), model order
  for (int base = 0; base < REMN; base += 1024) {
    int j = base + tid;
    int n = 0, pred = 0;
    if (j < REMN) { n = mn[NSPLIT + j]; pred = flags[n] ? 0 : 1; }
    sscan[tid] = pred;
    __syncthreads();
    for (int off = 1; off < 1024; off <<= 1) {
      int v = (tid >= off) ? sscan[tid - off] : 0;
      __syncthreads();
      sscan[tid] += v;
      __syncthreads();
    }
    int pos = sbase + sscan[tid] - pred;
    if (pred && pos < TARGET) idx_all[pos] = n;
    __syncthreads();
    if (tid == 0) sbase += sscan[1023];
    __syncthreads();
  }
}

// =====================================================================
// Kernel 7: y_dec[d] = sum_k x_dec[k] * W[d, idx_all[k]]  (tiny gather GEMV)
// =====================================================================
__global__ __launch_bounds__(256)
void ydec_gemv(const float* __restrict__ W, const float* __restrict__ xdec,
               const int* __restrict__ idx_all, float* __restrict__ out) {
  __shared__ float sx[TARGET];
  __shared__ int   sidx[TARGET];
  const int tid = threadIdx.x;
  for (int k = tid; k < TARGET; k += 256) { sx[k] = xdec[k]; sidx[k] = idx_all[k]; }
  __syncthreads();
  const int d = blockIdx.x * 256 + tid;
  const float* wr = W + (size_t)d * D_FF;
  float acc = 0.f;
  for (int k = 0; k < TARGET; ++k) acc += sx[k] * wr[sidx[k]];
  out[(size_t)S_TOK * D_MODEL + d] = acc;
}

// =====================================================================
extern "C" void kernel_launch(void* const* d_in, const int* in_sizes, int n_in,
                              void* d_out, int out_size, void* d_ws, size_t ws_size,
                              hipStream_t stream) {
  const float* x    = (const float*)d_in[0];   // (1,2048,11008)
  const float* W    = (const float*)d_in[1];   // (4096,11008)
  const float* xdec = (const float*)d_in[2];   // (1,1,4403)
  const int*   mn   = (const int*)d_in[3];     // (11008,)
  float* out = (float*)d_out;                  // (1,2049,4096) flat

  int* counts  = (int*)d_ws;
  int* flags   = counts + D_FF;
  int* idx_all = flags + D_FF;

  // big fp32 WMMA GEMM -> out rows [0, 2048)
  dim3 grid(D_MODEL / BN, S_TOK / BM);         // (16, 16)
  gemm_f32_wmma<<<grid, 256, 0, stream>>>(x, W, out);

  // selection pipeline
  init_ws<<<(D_FF + 255) / 256, 256, 0, stream>>>(counts, flags);
  topk_counts<<<S_TOK, 256, 0, stream>>>(x, counts);
  split_flags<<<(NSPLIT + 255) / 256, 256, 0, stream>>>(mn, flags);
  core_flags<<<1, 1024, 0, stream>>>(counts, flags);
  build_idx<<<1, 1024, 0, stream>>>(flags, mn, idx_all);

  // y_dec -> out row 2048
  ydec_gemv<<<D_MODEL / 256, 256, 0, stream>>>(W, xdec, idx_all, out);
}